// Dist_Conv2D_90855738180335
// MI455X (gfx1250) — compile-verified
//
#include <hip/hip_runtime.h>
#include <cstdint>
#include <cstddef>

#define IN_C   32
#define OUT_C  128
#define HH     64
#define WW     64
#define BB     32
#define TR     4                 // output rows per block
#define SLAB_ROWS (TR + 2)       // 6 input rows incl. halo
#define ROW_STRIDE 72            // floats per slab row (288B, 16B aligned)
#define COL0   4                 // x col 0 lives at position 4 (byte 16, B128-aligned)
                                 // pos 3 = left replicate dup, pos 68 = right replicate dup
#define SLAB_FLOATS (IN_C * SLAB_ROWS * ROW_STRIDE)   // 13824 floats = 55296 B

#if defined(__HIP_DEVICE_COMPILE__)
#if __has_builtin(__builtin_amdgcn_global_load_async_to_lds_b128) && \
    __has_builtin(__builtin_amdgcn_global_load_async_to_lds_b32)
#define ASYNC_LDS 1
#endif
#endif

typedef int v4i_t __attribute__((ext_vector_type(4)));
typedef __attribute__((address_space(1))) v4i_t* gv4p;   // global v4i*
typedef __attribute__((address_space(3))) v4i_t* lv4p;   // LDS v4i*
typedef __attribute__((address_space(1))) int*   gi32p;  // global int*
typedef __attribute__((address_space(3))) int*   li32p;  // LDS int*

__device__ __forceinline__ void async_copy_b128(const float* g, float* l) {
#if defined(ASYNC_LDS)
  __builtin_amdgcn_global_load_async_to_lds_b128(
      (gv4p)(v4i_t*)(void*)g, (lv4p)(v4i_t*)(void*)l, 0, 0);
#else
  float4 t = *(const float4*)g;
  *(float4*)l = t;
#endif
}

__device__ __forceinline__ void async_copy_b32(const float* g, float* l) {
#if defined(ASYNC_LDS)
  __builtin_amdgcn_global_load_async_to_lds_b32(
      (gi32p)(int*)(void*)g, (li32p)(int*)(void*)l, 0, 0);
#else
  *l = *g;
#endif
}

__device__ __forceinline__ void wait_async_all() {
#if defined(ASYNC_LDS)
#if __has_builtin(__builtin_amdgcn_s_wait_asynccnt)
  __builtin_amdgcn_s_wait_asynccnt(0);
#else
  asm volatile("s_wait_asynccnt 0" ::: "memory");
#endif
#endif
}

// ---------------------------------------------------------------------------
// Main kernel: one block = (batch b, 4 output rows). 128 threads, 2 pixels each.
// Defined FIRST so the disasm snippet shows this kernel.
// ---------------------------------------------------------------------------
__global__ __launch_bounds__(128)
void dist_conv2d_kernel(const float* __restrict__ x,
                        const float* __restrict__ wts,
                        const float* __restrict__ bias,
                        const int*   __restrict__ tab,
                        float*       __restrict__ out) {
  __shared__ float xs[SLAB_FLOATS];

  const int tid = threadIdx.x;
  const int b   = blockIdx.y;
  const int h0  = blockIdx.x * TR;
  const float* xb = x + (size_t)b * (IN_C * HH * WW);

  // ---- stage interior cols 0..63 at positions 4..67: 32ch x 6rows x 16 b128 ----
#pragma unroll
  for (int j = 0; j < 24; ++j) {
    int T   = tid + 128 * j;          // 0..3071
    int c   = T / 96;                 // 96 transfers per channel
    int rem = T - c * 96;
    int row = rem >> 4;
    int cg  = (rem & 15) << 2;        // starting column (multiple of 4)
    int grow = h0 - 1 + row;
    grow = grow < 0 ? 0 : (grow > HH - 1 ? HH - 1 : grow);
    async_copy_b128(xb + c * (HH * WW) + grow * WW + cg,
                    &xs[(c * SLAB_ROWS + row) * ROW_STRIDE + COL0 + cg]);
  }
  // ---- replicate halos: pos 3 = dup of col 0, pos 68 = dup of col 63 ----
  for (int t = tid; t < IN_C * SLAB_ROWS; t += 128) {
    int c   = t / SLAB_ROWS;
    int row = t - c * SLAB_ROWS;
    int grow = h0 - 1 + row;
    grow = grow < 0 ? 0 : (grow > HH - 1 ? HH - 1 : grow);
    const float* srow = xb + c * (HH * WW) + grow * WW;
    float* drow = &xs[(c * SLAB_ROWS + row) * ROW_STRIDE];
    async_copy_b32(srow,           drow + (COL0 - 1));   // left dup
    async_copy_b32(srow + (WW - 1), drow + (COL0 + WW)); // right dup
  }
  wait_async_all();
  __syncthreads();

  // ---- per-thread pixel geometry: rows r0 and r0+2 of the 4-row tile ----
  const int w  = tid & 63;
  const int r0 = tid >> 6;                      // 0 or 1
  const int Va = (r0 * ROW_STRIDE + w) * 4;     // divergent byte offsets into slab
  const int Vb = ((r0 + 2) * ROW_STRIDE + w) * 4;
  const char* xsb = (const char*)xs;

  const int pixA = (h0 + r0) * WW + w;          // divergent 32-bit store offsets
  const int pixB = pixA + 2 * WW;
  float* outBase = out + (size_t)b * (OUT_C * HH * WW);

#pragma unroll 8
  for (int o = 0; o < OUT_C; ++o) {
    const int   s0 = tab[3 * o + 0];            // uniform byte offsets (scalar loads)
    const int   s1 = tab[3 * o + 1];
    const int   s2 = tab[3 * o + 2];
    const float w0 = wts[3 * o + 0];
    const float w1 = wts[3 * o + 1];
    const float w2 = wts[3 * o + 2];
    const float bo = bias[o];

    const float a0 = *(const float*)(xsb + s0 + Va);
    const float a1 = *(const float*)(xsb + s1 + Va);
    const float a2 = *(const float*)(xsb + s2 + Va);
    const float c0 = *(const float*)(xsb + s0 + Vb);
    const float c1 = *(const float*)(xsb + s1 + Vb);
    const float c2 = *(const float*)(xsb + s2 + Vb);

    float dA = fmaxf(fmaxf(fabsf(w0 - a0), fabsf(w1 - a1)), fabsf(w2 - a2)) + bo;
    float dB = fmaxf(fmaxf(fabsf(w0 - c0), fabsf(w1 - c1)), fabsf(w2 - c2)) + bo;

    __builtin_nontemporal_store(dA, outBase + o * (HH * WW) + pixA);
    __builtin_nontemporal_store(dB, outBase + o * (HH * WW) + pixB);
  }
}

// ---------------------------------------------------------------------------
// Decode conn[] once per launch into pure LDS byte offsets:
//   tab[j] = ((c*6 + kh)*72 + 4 + (kw-1)) * 4
// Both replicate clamps are materialized in the slab, so no runtime selects.
// ---------------------------------------------------------------------------
__global__ void build_tab_kernel(const int* __restrict__ conn,
                                 int* __restrict__ tab) {
  int j = threadIdx.x;
  if (j < OUT_C * 3) {
    int idx = conn[j];
    int c  = idx / 9;
    int t  = idx - 9 * c;
    int kh = t / 3;
    int kw = t - 3 * kh;
    tab[j] = ((c * SLAB_ROWS + kh) * ROW_STRIDE + COL0 + (kw - 1)) * 4;
  }
}

extern "C" void kernel_launch(void* const* d_in, const int* in_sizes, int n_in,
                              void* d_out, int out_size, void* d_ws, size_t ws_size,
                              hipStream_t stream) {
  const float* x    = (const float*)d_in[0];   // [32,32,64,64]
  const float* wts  = (const float*)d_in[1];   // [128,3]
  const float* bias = (const float*)d_in[2];   // [128,1,1]
  const int*   conn = (const int*)d_in[3];     // [384]
  float* out = (float*)d_out;                  // [32,128,64,64]
  int* tab = (int*)d_ws;                       // 384 ints of scratch

  build_tab_kernel<<<1, OUT_C * 3, 0, stream>>>(conn, tab);

  dim3 grid(HH / TR, BB);                      // (16, 32)
  dist_conv2d_kernel<<<grid, 128, 0, stream>>>(x, wts, bias, tab, out);
}